// EoMTCriterion_54460185313556
// MI455X (gfx1250) — compile-verified
//
#include <hip/hip_runtime.h>

typedef __attribute__((ext_vector_type(2))) float v2f;
typedef __attribute__((ext_vector_type(8))) float v8f;

#define HW      262144      // 512*512
#define NQ      100         // num queries
#define NQPAD   112         // 7 tiles of 16
#define NGT     20          // num GT
#define MPAD    32          // 2 tiles of 16
#define CHUNK   256         // K pixels staged in LDS per iteration
#define NITER   2           // stages per block (1024 blocks total -> fills a big WGP array)
#define BLOCK_K (CHUNK*NITER)
#define LSTRIDE 33          // LDS row stride (dwords) to avoid bank conflicts
#define NO_OBJ  6

__global__ void zero_kernel(float* p, int n) {
  for (int i = blockIdx.x * blockDim.x + threadIdx.x; i < n; i += gridDim.x * blockDim.x)
    p[i] = 0.0f;
}

// Per K-chunk: accumulate PY[n][m] = p . y , LY[n][m] = (logp-log1mp) . y,
// SLN[n] = sum log(1-p).  Row 111 acts as ones-row (=> col sums of y),
// column 31 acts as ones-col (=> row sums of p).
__global__ __launch_bounds__(224) void accum_kernel(
    const float* __restrict__ msk, const int* __restrict__ gt,
    float* __restrict__ PY, float* __restrict__ LY, float* __restrict__ SLN) {
  __shared__ float ldsY[CHUNK * LSTRIDE];
  const int b      = blockIdx.y;
  const int kblock = blockIdx.x * BLOCK_K;
  const int tid    = threadIdx.x;
  const int lane   = tid & 31;
  const int rtile  = tid >> 5;        // 0..6 (7 waves = 7 row tiles)
  const int half   = lane >> 4;       // 0/1
  const int c      = lane & 15;
  const int arow   = rtile * 16 + c;  // A-operand row for this lane (both halves)
  const bool pad   = arow >= NQ;
  const float pfix = (arow == 111) ? 1.0f : 0.0f;
  const float* xrow = msk + (size_t)(b * NQ + (pad ? (NQ - 1) : arow)) * HW;

  v8f aPY0 = {}; v8f aPY1 = {}; v8f aLY0 = {}; v8f aLY1 = {};
  float sln = 0.0f;

  for (int it = 0; it < NITER; ++it) {
    const int k0 = kblock + it * CHUNK;
    __syncthreads();
    // stage y chunk: ldsY[k][m] for m<20; cols 20..30 = 0; col 31 = 1
    for (int q = tid; q < NGT * (CHUNK / 4); q += 224) {
      int m  = q / (CHUNK / 4);
      int k4 = (q % (CHUNK / 4)) * 4;
      const int4 v = *reinterpret_cast<const int4*>(gt + (size_t)(b * NGT + m) * HW + k0 + k4);
      ldsY[(k4 + 0) * LSTRIDE + m] = (float)v.x;
      ldsY[(k4 + 1) * LSTRIDE + m] = (float)v.y;
      ldsY[(k4 + 2) * LSTRIDE + m] = (float)v.z;
      ldsY[(k4 + 3) * LSTRIDE + m] = (float)v.w;
    }
    for (int q = tid; q < 12 * CHUNK; q += 224) {
      int k  = q / 12;
      int cc = 20 + (q % 12);
      ldsY[k * LSTRIDE + cc] = (cc == 31) ? 1.0f : 0.0f;
    }
    __syncthreads();

    for (int kk = 0; kk < CHUNK; kk += 4) {
      // A operand: lanes 0-15 hold K={kk,kk+1}, lanes 16-31 hold K={kk+2,kk+3}
      v2f xv = *reinterpret_cast<const v2f*>(xrow + k0 + kk + 2 * half);
      float p0, p1, ld0, ld1, ln0, ln1;
      {
        float x = xv[0];
        float e = __expf(-fabsf(x));
        float s = 1.0f / (1.0f + e);
        p0 = (x >= 0.0f) ? s : (1.0f - s);
        float lse = __logf(1.0f + e);
        float lp  = fmaxf(fminf(x, 0.0f) - lse, -100.0f);
        ln0 = fmaxf(fminf(-x, 0.0f) - lse, -100.0f);
        ld0 = lp - ln0;
      }
      {
        float x = xv[1];
        float e = __expf(-fabsf(x));
        float s = 1.0f / (1.0f + e);
        p1 = (x >= 0.0f) ? s : (1.0f - s);
        float lse = __logf(1.0f + e);
        float lp  = fmaxf(fminf(x, 0.0f) - lse, -100.0f);
        ln1 = fmaxf(fminf(-x, 0.0f) - lse, -100.0f);
        ld1 = lp - ln1;
      }
      p0  = pad ? pfix : p0;   p1  = pad ? pfix : p1;
      ld0 = pad ? 0.0f : ld0;  ld1 = pad ? 0.0f : ld1;
      sln += pad ? 0.0f : (ln0 + ln1);

      v2f aP = { p0, p1 };
      v2f aL = { ld0, ld1 };
      const float* yk = &ldsY[(kk + 2 * half) * LSTRIDE];
      v2f b0 = { yk[c],       yk[LSTRIDE + c] };
      v2f b1 = { yk[c + 16],  yk[LSTRIDE + c + 16] };

      aPY0 = __builtin_amdgcn_wmma_f32_16x16x4_f32(false, aP, false, b0, (short)0, aPY0, false, false);
      aPY1 = __builtin_amdgcn_wmma_f32_16x16x4_f32(false, aP, false, b1, (short)0, aPY1, false, false);
      aLY0 = __builtin_amdgcn_wmma_f32_16x16x4_f32(false, aL, false, b0, (short)0, aLY0, false, false);
      aLY1 = __builtin_amdgcn_wmma_f32_16x16x4_f32(false, aL, false, b1, (short)0, aLY1, false, false);
    }
  }

  // C/D layout: VGPR i -> M = i + 8*half, N = lane&15 (+16 for tile 1)
  {
    const int mrow = rtile * 16 + 8 * half;
    float* PYb = PY + (size_t)b * NQPAD * MPAD;
    float* LYb = LY + (size_t)b * NQPAD * MPAD;
#pragma unroll
    for (int i = 0; i < 8; ++i) {
      atomicAdd(&PYb[(mrow + i) * MPAD + c],      aPY0[i]);
      atomicAdd(&PYb[(mrow + i) * MPAD + c + 16], aPY1[i]);
      atomicAdd(&LYb[(mrow + i) * MPAD + c],      aLY0[i]);
      atomicAdd(&LYb[(mrow + i) * MPAD + c + 16], aLY1[i]);
    }
    atomicAdd(&SLN[b * NQPAD + arow], sln);
  }
}

__global__ void cost_kernel(const float* __restrict__ cls_logits,
                            const int* __restrict__ gt_cls,
                            const float* __restrict__ PY,
                            const float* __restrict__ LY,
                            const float* __restrict__ SLN,
                            float* __restrict__ COST,
                            float* __restrict__ BCE,
                            float* __restrict__ DICE) {
  const int b = blockIdx.x;
  for (int idx = threadIdx.x; idx < NQ * NGT; idx += blockDim.x) {
    int n = idx / NGT, m = idx % NGT;
    const float* lg = cls_logits + (b * NQ + n) * 7;
    float mx = lg[0];
    for (int k = 1; k < 7; ++k) mx = fmaxf(mx, lg[k]);
    float se = 0.0f;
    for (int k = 0; k < 7; ++k) se += __expf(lg[k] - mx);
    int g = gt_cls[b * NGT + m];
    float prob = __expf(lg[g] - mx) / se;

    float ly  = LY[((size_t)b * NQPAD + n) * MPAD + m];
    float sln = SLN[b * NQPAD + n];
    float bce = -(ly + sln) * (1.0f / (float)HW);
    float py  = PY[((size_t)b * NQPAD + n) * MPAD + m];
    float sp  = PY[((size_t)b * NQPAD + n) * MPAD + 31];
    float sy  = PY[((size_t)b * NQPAD + 111) * MPAD + m];
    float dice = 1.0f - (2.0f * py + 1.0f) / (sp + sy + 1.0f);

    COST[b * NQ * NGT + idx] = 2.0f * (-prob) + 5.0f * bce + 5.0f * dice;
    BCE[b * NQ * NGT + idx]  = bce;
    DICE[b * NQ * NGT + idx] = dice;
  }
}

// Wave-parallel Jonker-Volgenant LSA on transposed cost [20 x 100],
// matching the reference (first-minimum tie-break). One wave per batch.
__global__ void hungarian_kernel(const float* __restrict__ COST, int* __restrict__ MATCH) {
  const int b = blockIdx.x;
  const int lane = threadIdx.x;  // 32 threads
  __shared__ float costT[NGT * NQ];
  __shared__ float u[NGT + 1], vv[NQ + 1], minv[NQ + 1];
  __shared__ int p[NQ + 1], way[NQ + 1], used[NQ + 1];
  __shared__ float s_delta;
  __shared__ int s_j1, s_j0, s_i0;

  for (int idx = lane; idx < NGT * NQ; idx += 32) {
    int i = idx / NQ, j = idx % NQ;
    costT[idx] = COST[(b * NQ + j) * NGT + i];
  }
  for (int j = lane; j <= NQ; j += 32) { vv[j] = 0.0f; p[j] = 0; way[j] = 0; }
  for (int i = lane; i <= NGT; i += 32) u[i] = 0.0f;
  __syncthreads();

  for (int i = 1; i <= NGT; ++i) {
    if (lane == 0) { p[0] = i; s_j0 = 0; }
    for (int j = lane; j <= NQ; j += 32) { minv[j] = 1e30f; used[j] = 0; }
    __syncthreads();
    while (true) {
      if (lane == 0) { used[s_j0] = 1; s_i0 = p[s_j0]; }
      __syncthreads();
      const int j0 = s_j0, i0 = s_i0;
      const float ui0 = u[i0];
      float best = 1e30f; int bestj = NQ + 1;
      for (int j = 1 + lane; j <= NQ; j += 32) {
        if (!used[j]) {
          float cand = costT[(i0 - 1) * NQ + (j - 1)] - ui0 - vv[j];
          if (cand < minv[j]) { minv[j] = cand; way[j] = j0; }
        }
        float mv = used[j] ? 1e30f : minv[j];
        if (mv < best) { best = mv; bestj = j; }
      }
      for (int off = 16; off > 0; off >>= 1) {
        float ov = __shfl_xor(best, off, 32);
        int   oj = __shfl_xor(bestj, off, 32);
        if (ov < best || (ov == best && oj < bestj)) { best = ov; bestj = oj; }
      }
      if (lane == 0) { s_delta = best; s_j1 = bestj; }
      __syncthreads();
      const float delta = s_delta; const int j1 = s_j1;
      for (int j = lane; j <= NQ; j += 32) {
        if (used[j]) { u[p[j]] += delta; vv[j] -= delta; }
        else         { minv[j] -= delta; }
      }
      __syncthreads();
      if (lane == 0) s_j0 = j1;
      __syncthreads();
      if (p[j1] == 0) break;
    }
    if (lane == 0) {
      int j0 = s_j0;
      while (j0) { int jn = way[j0]; p[j0] = p[jn]; j0 = jn; }
    }
    __syncthreads();
  }
  for (int j = 1 + lane; j <= NQ; j += 32)
    MATCH[b * NQ + (j - 1)] = p[j] - 1;   // gt index matched to pred, or -1
}

__global__ void finalize_kernel(const float* __restrict__ cls_logits,
                                const int* __restrict__ gt_cls,
                                const float* __restrict__ BCE,
                                const float* __restrict__ DICE,
                                const int* __restrict__ MATCH,
                                float* __restrict__ out) {
  if (threadIdx.x != 0) return;
  float tc = 0.0f, tm = 0.0f, td = 0.0f;
  for (int b = 0; b < 2; ++b) {
    float wn = 0.0f, wsum = 0.0f, bs = 0.0f, ds = 0.0f;
    for (int n = 0; n < NQ; ++n) {
      int g = MATCH[b * NQ + n];
      int target = (g >= 0) ? gt_cls[b * NGT + g] : NO_OBJ;
      const float* lg = cls_logits + (b * NQ + n) * 7;
      float mx = lg[0];
      for (int k = 1; k < 7; ++k) mx = fmaxf(mx, lg[k]);
      float se = 0.0f;
      for (int k = 0; k < 7; ++k) se += __expf(lg[k] - mx);
      float nll = (mx + __logf(se)) - lg[target];
      float w = (target == NO_OBJ) ? 0.1f : 1.0f;
      wn += w * nll; wsum += w;
      if (g >= 0) { bs += BCE[(b * NQ + n) * NGT + g]; ds += DICE[(b * NQ + n) * NGT + g]; }
    }
    tc += wn / wsum;
    tm += bs * (1.0f / (float)NGT);
    td += ds * (1.0f / (float)NGT);
  }
  tc *= 0.5f; tm *= 0.5f; td *= 0.5f;
  out[0] = tc; out[1] = tm; out[2] = td;
  out[3] = 2.0f * tc + 5.0f * tm + 5.0f * td;
}

extern "C" void kernel_launch(void* const* d_in, const int* in_sizes, int n_in,
                              void* d_out, int out_size, void* d_ws, size_t ws_size,
                              hipStream_t stream) {
  const float* cls_logits = (const float*)d_in[0];  // [2,100,7]
  const float* msk_logits = (const float*)d_in[1];  // [2,100,512,512]
  const int*   gt_classes = (const int*)d_in[2];    // [2,20]
  const int*   gt_masks   = (const int*)d_in[3];    // [2,20,512,512]
  float* out = (float*)d_out;

  float* PY   = (float*)d_ws;                 // 2*112*32
  float* LY   = PY + 2 * NQPAD * MPAD;        // 2*112*32
  float* SLN  = LY + 2 * NQPAD * MPAD;        // 2*112
  float* COST = SLN + 2 * NQPAD;              // 2*100*20
  float* BCE  = COST + 2 * NQ * NGT;
  float* DICE = BCE + 2 * NQ * NGT;
  int*   MATCH = (int*)(DICE + 2 * NQ * NGT); // 2*100

  zero_kernel<<<16, 256, 0, stream>>>(PY, 2 * NQPAD * MPAD * 2 + 2 * NQPAD);
  dim3 grid(HW / BLOCK_K, 2);
  accum_kernel<<<grid, 224, 0, stream>>>(msk_logits, gt_masks, PY, LY, SLN);
  cost_kernel<<<2, 128, 0, stream>>>(cls_logits, gt_classes, PY, LY, SLN, COST, BCE, DICE);
  hungarian_kernel<<<2, 32, 0, stream>>>(COST, MATCH);
  finalize_kernel<<<1, 32, 0, stream>>>(cls_logits, gt_classes, BCE, DICE, MATCH, out);
}